// GraphConsis_72816875537004
// MI455X (gfx1250) — compile-verified
//
#include <hip/hip_runtime.h>
#include <math.h>

typedef float v2f __attribute__((ext_vector_type(2)));
typedef float v8f __attribute__((ext_vector_type(8)));

// Problem constants (from the reference)
#define DEVF   34   // device_feats row width (32 dense + 2 cat)
#define COMF   49   // combin_feats row width (48 dense + 1 cat)
#define DDEV   80   // assembled device feature dim: 16 ctx + 32 dense + 16 + 16
#define DCOM   64   // assembled combin feature dim: 48 dense + 16
#define NOUT   64   // u/v dim
#define SA_STRIDE 81
#define SV_STRIDE 65

// Assemble one element of the 80-dim device feature vector for device g
// (used only for the single query row dfb[b]; the hot gather is vectorized).
__device__ __forceinline__ float dev_feat(const float* __restrict__ df,
                                          const float* __restrict__ ctx,
                                          const float* __restrict__ e0,
                                          const float* __restrict__ e1,
                                          int g, int j) {
    if (j < 16) return ctx[g * 16 + j];
    if (j < 48) return df[g * DEVF + (j - 16)];
    if (j < 64) { int c0 = (int)df[g * DEVF + 32]; return e0[c0 * 16 + (j - 48)]; }
    int c1 = (int)df[g * DEVF + 33]; return e1[c1 * 16 + (j - 64)];
}

__global__ __launch_bounds__(128)
void graphconsis_kernel(const float* __restrict__ combin_feats,
                        const float* __restrict__ device_feats,
                        const float* __restrict__ context_embed,
                        const float* __restrict__ dev_emb0,
                        const float* __restrict__ dev_emb1,
                        const float* __restrict__ com_emb0,
                        const float* __restrict__ Wc, const float* __restrict__ bc,
                        const float* __restrict__ Wd, const float* __restrict__ bd,
                        const float* __restrict__ W1, const float* __restrict__ b1,
                        const float* __restrict__ W2, const float* __restrict__ b2,
                        const float* __restrict__ W3, const float* __restrict__ b3,
                        const int* __restrict__ combin_idx,
                        const int* __restrict__ device_idx,
                        const int* __restrict__ neibr_idx,
                        const int* __restrict__ neibr_len,
                        float* __restrict__ out, int M)
{
    __shared__ float sA[16 * SA_STRIDE];   // gathered neighbor tile, 16 x 80
    __shared__ float sV[16 * SV_STRIDE];   // v tile, 16 x 64
    __shared__ float sU[NOUT];             // u[b]
    __shared__ float sDfb[DDEV];           // dfb[b]
    __shared__ float sScore[16];
    __shared__ float sAggU[NOUT];
    __shared__ float sAggN[NOUT];
    __shared__ float sRed[16];
    __shared__ float sH1[64];
    __shared__ float sH2[32];

    const int b    = blockIdx.x;
    const int t    = threadIdx.x;      // 0..127
    const int wave = t >> 5;           // 0..3 -> owns output columns [wave*16, wave*16+16)
    const int lane = t & 31;
    const int len  = neibr_len[b];

    // ---- u = cf @ Wc.T + bc  (threads 0..63, one output each)
    if (t < NOUT) {
        const int ci   = combin_idx[b];
        const int ccat = (int)combin_feats[ci * COMF + 48];
        const float* wrow = Wc + t * DCOM;
        float acc = bc[t];
        for (int k = 0; k < 48; ++k) acc = fmaf(combin_feats[ci * COMF + k], wrow[k], acc);
        for (int k = 0; k < 16; ++k) acc = fmaf(com_emb0[ccat * 16 + k], wrow[48 + k], acc);
        sU[t] = acc;
    }
    // ---- dfb[b] : assembled 80-dim feature of the query device
    if (t < DDEV) sDfb[t] = dev_feat(device_feats, context_embed, dev_emb0, dev_emb1,
                                     device_idx[b], t);

    // ---- Preload this wave's loop-invariant B fragments of Wd.T (80x16 slice).
    // B layout (4x16 f32, mirrors documented A 16x4): VGPR0 = {K=0 | K=2},
    // VGPR1 = {K=1 | K=3}; lanes 0-15 / 16-31 hold N=0..15 of the slice.
    const int n0 = wave * 16;
    const int nn = n0 + (lane & 15);
    const int kbase = (lane < 16) ? 0 : 2;
    v2f bfrag[20];
    #pragma unroll
    for (int kk = 0; kk < 20; ++kk) {
        bfrag[kk].x = Wd[nn * DDEV + kk * 4 + kbase];
        bfrag[kk].y = Wd[nn * DDEV + kk * 4 + kbase + 1];
    }
    const float bias = bd[nn];

    float ssum_part = 0.0f;  // threads with t<64 && (t&3)==0
    float aggU = 0.0f;       // threads 0..63: column t of  sum_m s_m * v_m
    float aggN = 0.0f;       // threads 64..127: column t-64 of sum_{m<len} v_m

    const int ntiles = M >> 4;
    const int grow = t >> 3;   // gather: 8 threads per row
    const int gcol = t & 7;    // chunk lane within row (5 float2 chunks each)
    const int* __restrict__ idx_base = neibr_idx + b * M + grow;

    // ================= 3-stage gather pipeline =================
    // Entering iteration `tile`:
    //   pre[5]        : features of `tile`        (ready to commit to LDS)
    //   gF, c0F, c1F  : index + cats of tile+1    (addresses ready for feature loads)
    //   gC            : index of tile+2           (address ready for cat loads)
    float2 pre[5];
    int gF = 0, c0F = 0, c1F = 0, gC = 0;
    {
        // prologue: fill the pipeline for tiles 0,1,2
        const int g0  = idx_base[0];
        const int c00 = (int)device_feats[g0 * DEVF + 32];
        const int c01 = (int)device_feats[g0 * DEVF + 33];
        const int o = 2 * gcol;
        pre[0] = *(const float2*)(context_embed + g0 * 16 + o);
        pre[1] = *(const float2*)(device_feats + g0 * DEVF + o);
        pre[2] = *(const float2*)(device_feats + g0 * DEVF + 16 + o);
        pre[3] = *(const float2*)(dev_emb0 + c00 * 16 + o);
        pre[4] = *(const float2*)(dev_emb1 + c01 * 16 + o);
        if (ntiles > 1) {
            gF  = idx_base[16];
            c0F = (int)device_feats[gF * DEVF + 32];
            c1F = (int)device_feats[gF * DEVF + 33];
        }
        if (ntiles > 2) gC = idx_base[32];
    }

    for (int tile = 0; tile < ntiles; ++tile) {
        // ---- commit gathered features of `tile` into LDS (waits on its loads)
        {
            float* dstrow = sA + grow * SA_STRIDE;
            const int o = 2 * gcol;
            dstrow[o]      = pre[0].x;  dstrow[o + 1]      = pre[0].y;
            dstrow[16 + o] = pre[1].x;  dstrow[16 + o + 1] = pre[1].y;
            dstrow[32 + o] = pre[2].x;  dstrow[32 + o + 1] = pre[2].y;
            dstrow[48 + o] = pre[3].x;  dstrow[48 + o + 1] = pre[3].y;
            dstrow[64 + o] = pre[4].x;  dstrow[64 + o + 1] = pre[4].y;
        }
        __syncthreads();

        // ---- issue next-stage loads; all addresses are register-resident,
        //      results are not consumed until the next iteration's commit.
        if (tile + 1 < ntiles) {
            const int o = 2 * gcol;
            pre[0] = *(const float2*)(context_embed + gF * 16 + o);
            pre[1] = *(const float2*)(device_feats + gF * DEVF + o);
            pre[2] = *(const float2*)(device_feats + gF * DEVF + 16 + o);
            pre[3] = *(const float2*)(dev_emb0 + c0F * 16 + o);
            pre[4] = *(const float2*)(dev_emb1 + c1F * 16 + o);
        }
        if (tile + 2 < ntiles) {
            c0F = (int)device_feats[gC * DEVF + 32];
            c1F = (int)device_feats[gC * DEVF + 33];
            gF  = gC;
        }
        if (tile + 3 < ntiles) gC = idx_base[(tile + 3) * 16];

        // ---- v tile = A(16x80) @ Wd.T(80x16slice) + bd   via WMMA f32 16x16x4
        v8f c;
        #pragma unroll
        for (int i = 0; i < 8; ++i) c[i] = bias;
        const int arow = lane & 15;
        const int koff = (lane < 16) ? 0 : 2;
        #pragma unroll
        for (int kk = 0; kk < 20; ++kk) {
            v2f a;
            a.x = sA[arow * SA_STRIDE + kk * 4 + koff];
            a.y = sA[arow * SA_STRIDE + kk * 4 + koff + 1];
            c = __builtin_amdgcn_wmma_f32_16x16x4_f32(
                    false, a, false, bfrag[kk], (short)0, c, false, false);
        }
        // C layout: VGPR i -> (M = i + (lane<16?0:8), N = n0 + lane%16)
        {
            const int mbase = (lane < 16) ? 0 : 8;
            #pragma unroll
            for (int i = 0; i < 8; ++i)
                sV[(mbase + i) * SV_STRIDE + nn] = c[i];
        }
        __syncthreads();

        // ---- scores: 4 threads per neighbor row, wave32 shuffle reduce
        if (t < 64) {
            const int row  = t >> 2;
            const int part = t & 3;
            const float* vrow = sV + row * SV_STRIDE + part * 16;
            const float* urow = sU + part * 16;
            float d2 = 0.0f;
            #pragma unroll
            for (int i = 0; i < 16; ++i) {
                float d = urow[i] - vrow[i] + 1e-6f;
                d2 = fmaf(d, d, d2);
            }
            d2 += __shfl_xor(d2, 1);
            d2 += __shfl_xor(d2, 2);
            if (part == 0) {
                float s = __expf(-sqrtf(d2));
                const int mg = tile * 16 + row;
                s = (s > 0.001f && mg < len) ? s : 0.0f;
                sScore[row] = s;
                ssum_part += s;
            }
        }
        __syncthreads();

        // ---- accumulate weighted & uniform aggregates
        if (t < NOUT) {
            float a = 0.0f;
            #pragma unroll
            for (int m = 0; m < 16; ++m)
                a = fmaf(sScore[m], sV[m * SV_STRIDE + t], a);
            aggU += a;
        } else {
            const int n = t - NOUT;
            int rem = len - tile * 16;
            int vm = rem < 0 ? 0 : (rem > 16 ? 16 : rem);
            float a = 0.0f;
            for (int m = 0; m < vm; ++m) a += sV[m * SV_STRIDE + n];
            aggN += a;
        }
        __syncthreads();
    }

    // ---- reduce ssum, finalize agg
    if (t < 64 && (t & 3) == 0) sRed[t >> 2] = ssum_part;
    if (t < NOUT) sAggU[t] = aggU;
    if (t >= NOUT) sAggN[t - NOUT] = aggN;
    __syncthreads();
    if (t == 0) {
        float s = 0.0f;
        for (int i = 0; i < 16; ++i) s += sRed[i];
        sRed[0] = s;
    }
    __syncthreads();
    const float ssum = sRed[0];
    if (t < NOUT) {
        float a = (ssum > 0.0f) ? sAggU[t] / fmaxf(ssum, 1e-20f)
                                : sAggN[t] / (float)len;
        sAggU[t] = a;   // sAggU now holds agg[b]
    }
    __syncthreads();

    // ---- MLP head: x = [dfb(80), agg(64)] -> 64 -> 32 -> 1
    if (t < 64) {
        const float* w = W1 + t * (DDEV + NOUT);
        float acc = b1[t];
        for (int k = 0; k < DDEV; ++k) acc = fmaf(sDfb[k], w[k], acc);
        for (int k = 0; k < NOUT; ++k) acc = fmaf(sAggU[k], w[DDEV + k], acc);
        sH1[t] = fmaxf(acc, 0.0f);
    }
    __syncthreads();
    if (t < 32) {
        const float* w = W2 + t * 64;
        float acc = b2[t];
        for (int k = 0; k < 64; ++k) acc = fmaf(sH1[k], w[k], acc);
        sH2[t] = fmaxf(acc, 0.0f);
    }
    __syncthreads();
    if (t == 0) {
        float acc = b3[0];
        for (int k = 0; k < 32; ++k) acc = fmaf(sH2[k], W3[k], acc);
        out[b] = 1.0f / (1.0f + __expf(-acc));
    }
}

extern "C" void kernel_launch(void* const* d_in, const int* in_sizes, int n_in,
                              void* d_out, int out_size, void* d_ws, size_t ws_size,
                              hipStream_t stream) {
    const float* combin_feats  = (const float*)d_in[0];
    const float* device_feats  = (const float*)d_in[1];
    const float* context_embed = (const float*)d_in[2];
    const float* dev_emb0      = (const float*)d_in[3];
    const float* dev_emb1      = (const float*)d_in[4];
    const float* com_emb0      = (const float*)d_in[5];
    const float* Wc = (const float*)d_in[6];
    const float* bc = (const float*)d_in[7];
    const float* Wd = (const float*)d_in[8];
    const float* bd = (const float*)d_in[9];
    const float* W1 = (const float*)d_in[10];
    const float* b1 = (const float*)d_in[11];
    const float* W2 = (const float*)d_in[12];
    const float* b2 = (const float*)d_in[13];
    const float* W3 = (const float*)d_in[14];
    const float* b3 = (const float*)d_in[15];
    const int* combin_idx = (const int*)d_in[16];
    const int* device_idx = (const int*)d_in[17];
    const int* neibr_idx  = (const int*)d_in[18];
    const int* neibr_len  = (const int*)d_in[19];

    const int B = in_sizes[16];            // 1024
    const int M = in_sizes[18] / B;        // 1024

    graphconsis_kernel<<<dim3(B), dim3(128), 0, stream>>>(
        combin_feats, device_feats, context_embed, dev_emb0, dev_emb1, com_emb0,
        Wc, bc, Wd, bd, W1, b1, W2, b2, W3, b3,
        combin_idx, device_idx, neibr_idx, neibr_len,
        (float*)d_out, M);
}